// SpatialTransformer_38585986187350
// MI455X (gfx1250) — compile-verified
//
#include <hip/hip_runtime.h>
#include <hip/hip_bf16.h>

// ---------------------------------------------------------------------------
// SpatialTransformer bilinear gather for MI455X (gfx1250).
// Gather-bound kernel. 4 sample points per block: the Tensor Data Mover
// gathers 16 rows (4 points x 4 corners) of 2KB per batch with ONE
// tensor_load_to_lds descriptor (gather mode, 16-bit row indices = the
// gather-mode maximum), fenced by TENSORcnt; waves then blend from LDS with
// the 4 bilinear weights + mask product and store coalesced float4s.
// ---------------------------------------------------------------------------

constexpr int kOutH = 177;
constexpr int kOutW = 152;
constexpr int kP    = kOutH * kOutW;   // 26904 sample points (divisible by 4)
constexpr int kH    = 192;
constexpr int kW    = 256;
constexpr int kC    = 512;             // channels (= TDM row length)
constexpr int kC4   = kC / 4;          // 128 float4 per image row
constexpr int kHW   = kH * kW;         // 49152 rows per batch (fits 16-bit idx)
constexpr int kPPB  = 4;               // points per block -> 16 gather rows

typedef unsigned int u32;
typedef unsigned long long u64;
typedef u32 u32x4 __attribute__((ext_vector_type(4)));
typedef u32 u32x8 __attribute__((ext_vector_type(8)));

// Force a (wave-uniform-in-fact) value into an SGPR so inline-asm "s"
// constraints get scalar registers.
__device__ __forceinline__ u32 rfl(u32 x) {
  return (u32)__builtin_amdgcn_readfirstlane((int)x);
}

__device__ __forceinline__ void st_wait_tensor() {
#if __has_builtin(__builtin_amdgcn_s_wait_tensorcnt)
  __builtin_amdgcn_s_wait_tensorcnt(0);
#else
  asm volatile("s_wait_tensorcnt 0x0" ::: "memory");
#endif
}

struct PointGeom {
  u32 ia, ib, ic, id;          // gathered row indices (a,b,c,d corners)
  float wa, wb, wc, wd;        // bilinear weights (from clipped corners)
};

__device__ __forceinline__ PointGeom st_geom(const float* __restrict__ grid,
                                             int p) {
  const float gx = grid[p];
  const float gy = grid[kP + p];
  const float xf = 0.5f * (gx + 1.0f) * (float)kW;  // uses W, not W-1 (per ref)
  const float yf = 0.5f * (gy + 1.0f) * (float)kH;
  int x0 = (int)floorf(xf);
  int y0 = (int)floorf(yf);
  int x1 = x0 + 1;
  int y1 = y0 + 1;
  x0 = min(max(x0, 0), kW - 1);
  x1 = min(max(x1, 0), kW - 1);
  y0 = min(max(y0, 0), kH - 1);
  y1 = min(max(y1, 0), kH - 1);
  const float x0f = (float)x0, x1f = (float)x1;
  const float y0f = (float)y0, y1f = (float)y1;
  PointGeom g;
  g.wa = (x1f - xf) * (y1f - yf);
  g.wb = (x1f - xf) * (yf - y0f);
  g.wc = (xf - x0f) * (y1f - yf);
  g.wd = (xf - x0f) * (yf - y0f);
  g.ia = (u32)(y0 * kW + x0);
  g.ib = (u32)(y1 * kW + x0);
  g.ic = (u32)(y0 * kW + x1);
  g.id = (u32)(y1 * kW + x1);
  return g;
}

// One block per 4 sample points. 256 threads = 8 waves (wave32).
// Phase 1: wave 0 -> batch-0 descriptor, wave 1 -> batch-1 descriptor; each
//          TDM gather moves 16 rows x 2KB = 32KB into LDS.
// Phase 2: 64 threads per point blend 2 batches x 512 channels, store float4.
__global__ __launch_bounds__(256)
void st_gather_tdm(const float* __restrict__ image,
                   const float* __restrict__ grid,
                   const float* __restrict__ gp_mask,
                   const float* __restrict__ norm_mask,
                   float* __restrict__ out) {
  __shared__ float4 smem[2 * 16 * kC4];  // 64 KB: [batch][point*4+corner][c]

  const int t  = threadIdx.x;
  const int p0 = blockIdx.x * kPPB;      // first point of this block

  // ---- phase 1: two TDM gather descriptors (one per batch) ---------------
  const int wv = t >> 5;                 // wave index 0..7
  if (wv < 2) {                          // wave 0 -> batch 0, wave 1 -> batch 1
    // 16 packed 16-bit row indices: point-major, corner order a,b,c,d.
    u32 iw[8];
#pragma unroll
    for (int q = 0; q < kPPB; ++q) {
      const PointGeom g = st_geom(grid, p0 + q);
      iw[2 * q + 0] = g.ia | (g.ib << 16);
      iw[2 * q + 1] = g.ic | (g.id << 16);
    }

    // LDS destination: wave-relative byte offset of this batch's 16-row block.
    const u32 lds_off = (u32)(size_t)(const void*)&smem[0] + (u32)wv * 32768u;
    // Global base of this batch's [kHW x kC] fp32 tensor.
    const u64 ga = (u64)(size_t)image + (u64)wv * (u64)kHW * (u64)kC * 4ull;

    // D# group 0: count=1, gather_mode=1 (16-bit indices), lds_addr,
    //             global_addr[56:0], type=2 ("image").
    const u32x4 g0 = {
        0x80000001u,                                    // count=1 | gather<<31
        rfl(lds_off),                                   // lds_addr
        rfl((u32)ga),                                   // global_addr[31:0]
        rfl(((u32)(ga >> 32) & 0x01FFFFFFu) | 0x80000000u) // [56:32] | type=2
    };
    // D# group 1: data_size=4B; tensor_dim0=512; tensor_dim1=49152;
    //             tile_dim0=512; tile_dim1=16 valid indices;
    //             tensor_dim0_stride=512; no mask/barrier/iterate/pad.
    const u32x8 g1 = {
        0x00020000u,                    // workgroup_mask=0, data_size=2 (4B)
        ((u32)kC & 0xFFFFu) << 16,      // tensor_dim0[15:0] @ bits 63:48
        ((u32)kHW & 0xFFFFu) << 16,     // dim0 hi=0 | tensor_dim1[15:0]
        ((u32)kC) << 16,                // dim1 hi=0 | tile_dim0=512
        16u,                            // tile_dim1 = 16 gather indices (max)
        (u32)kC,                        // tensor_dim0_stride = 512
        0u,                             // stride hi / dim1_stride (ignored)
        0u
    };
    // D# groups 2/3: packed 16-bit row indices 0..7 and 8..15.
    const u32x4 g2 = { rfl(iw[0]), rfl(iw[1]), rfl(iw[2]), rfl(iw[3]) };
    const u32x4 g3 = { rfl(iw[4]), rfl(iw[5]), rfl(iw[6]), rfl(iw[7]) };

    asm volatile("tensor_load_to_lds %0, %1, %2, %3"
                 :: "s"(g0), "s"(g1), "s"(g2), "s"(g3)
                 : "memory");
    st_wait_tensor();                  // this wave's TDM transfer complete
  }
  __syncthreads();                     // release all waves to read LDS

  // ---- phase 2: blend + store -------------------------------------------
  const int q   = t >> 6;              // point within block 0..3 (wave-uniform)
  const int p   = p0 + q;
  const int col = t & 63;              // float4 column group

  const PointGeom g = st_geom(grid, p);          // scalarized per-wave
  const float m = gp_mask[p] * norm_mask[p];

  float4* out4 = (float4*)out;
#pragma unroll
  for (int b = 0; b < 2; ++b) {
#pragma unroll
    for (int h = 0; h < 2; ++h) {
      const int j   = col + 64 * h;              // 0..127
      const int row = (b * 16 + q * 4) * kC4 + j;
      const float4 pa = smem[row + 0 * kC4];
      const float4 pb = smem[row + 1 * kC4];
      const float4 pc = smem[row + 2 * kC4];
      const float4 pd = smem[row + 3 * kC4];
      float4 v;
      v.x = m * fmaf(g.wa, pa.x, fmaf(g.wb, pb.x, fmaf(g.wc, pc.x, g.wd * pd.x)));
      v.y = m * fmaf(g.wa, pa.y, fmaf(g.wb, pb.y, fmaf(g.wc, pc.y, g.wd * pd.y)));
      v.z = m * fmaf(g.wa, pa.z, fmaf(g.wb, pb.z, fmaf(g.wc, pc.z, g.wd * pd.z)));
      v.w = m * fmaf(g.wa, pa.w, fmaf(g.wb, pb.w, fmaf(g.wc, pc.w, g.wd * pd.w)));
      out4[((size_t)b * kP + (size_t)p) * kC4 + j] = v;
    }
  }
}

extern "C" void kernel_launch(void* const* d_in, const int* in_sizes, int n_in,
                              void* d_out, int out_size, void* d_ws, size_t ws_size,
                              hipStream_t stream) {
  const float* image     = (const float*)d_in[0];
  const float* grid      = (const float*)d_in[1];
  const float* gp_mask   = (const float*)d_in[2];
  const float* norm_mask = (const float*)d_in[3];
  float* out             = (float*)d_out;

  st_gather_tdm<<<dim3(kP / kPPB), dim3(256), 0, stream>>>(image, grid, gp_mask,
                                                           norm_mask, out);
}